// GraphConvolutionLayer_10591389352061
// MI455X (gfx1250) — compile-verified
//
#include <hip/hip_runtime.h>
#include <hip/hip_bf16.h>

#define IN_FEATS 128
#define OUT_FEATS 128

typedef __attribute__((ext_vector_type(2))) float v2f;
typedef __attribute__((ext_vector_type(8))) float v8f;

// ---------------------------------------------------------------------------
// Kernel 1: edge scatter.  One 32-lane wave per edge; the edge id is forced
// into an SGPR via readfirstlane so src[e]/dst[e] become scalar (SMEM) loads
// and the vector pipe only carries the b128 feature gather + 4 f32 atomics.
// agg (25.6 MB) is L2-resident (192 MB) so atomics RMW near-core.
// ---------------------------------------------------------------------------
__global__ void gc_scatter_kernel(const float* __restrict__ feats,
                                  const int* __restrict__ src,
                                  const int* __restrict__ dst,
                                  float* __restrict__ agg,
                                  int n_edges) {
    // 8 waves per block, one edge per wave (wave-uniform -> scalarized)
    const int wave = __builtin_amdgcn_readfirstlane((int)(threadIdx.x >> 5));
    const int e    = blockIdx.x * 8 + wave;
    if (e >= n_edges) return;
    const int lane = threadIdx.x & 31;

    const int s = src[e];   // SGPR address -> s_load_b32
    const int d = dst[e];

    const float4* frow = (const float4*)(feats + (size_t)s * IN_FEATS);
    float4 v = frow[lane];  // 32 lanes x 16B = full 512B row, coalesced

    float* out = agg + (size_t)d * IN_FEATS + lane * 4;
    atomicAdd(out + 0, v.x);
    atomicAdd(out + 1, v.y);
    atomicAdd(out + 2, v.z);
    atomicAdd(out + 3, v.w);
}

// ---------------------------------------------------------------------------
// Kernel 2: h = relu(agg @ W^T + bias) via V_WMMA_F32_16X16X4_F32 (exact f32,
// matches the fp32 reference).  Block = 256 threads = 8 waves; block b owns
// M-tile b (16 node rows), wave w owns N-tile w (16 output features).
// K-loop: 32 chained 16x16x4 WMMAs over the 128-deep reduction, unrolled 8x
// so 16 b64 fragment loads stay in flight ahead of the WMMA chain.
//
// Per ISA VGPR layouts (CDNA5 ISA):
//   A 16x4 f32 : lanes 0-15 -> (M=lane, K=k..k+1), lanes 16-31 -> K=k+2..k+3
//   B 4x16 f32 : mirrored, N = lane%16; B(k,n) = weight[n*128 + k] (row-major
//                weight[OUT][IN] is already B^T) -> contiguous float2 loads.
//   C/D 16x16  : VGPR i -> row i + 8*(lane>=16), col lane%16
// ---------------------------------------------------------------------------
__global__ void gc_gemm_relu_kernel(const float* __restrict__ agg,
                                    const float* __restrict__ weight,
                                    const float* __restrict__ bias,
                                    float* __restrict__ out) {
    const int wave = threadIdx.x >> 5;   // 0..7  -> N tile
    const int lane = threadIdx.x & 31;
    const int half = lane >> 4;          // 0: K pair k,k+1   1: K pair k+2,k+3
    const int l    = lane & 15;

    const int m0 = blockIdx.x * 16;
    const int n0 = wave * 16;

    const float* arow = agg    + (size_t)(m0 + l) * IN_FEATS + 2 * half;
    const float* brow = weight + (size_t)(n0 + l) * IN_FEATS + 2 * half;

    v8f c = {};
#pragma unroll 8
    for (int kk = 0; kk < IN_FEATS; kk += 4) {
        v2f a = *(const v2f*)(arow + kk);   // 8B aligned
        v2f b = *(const v2f*)(brow + kk);
        c = __builtin_amdgcn_wmma_f32_16x16x4_f32(
                /*neg_a=*/false, a, /*neg_b=*/false, b,
                /*c_mod=*/(short)0, c, /*reuse_a=*/false, /*reuse_b=*/false);
    }

    const float bv = bias[n0 + l];
    float* obase = out + (size_t)m0 * OUT_FEATS + n0 + l;
#pragma unroll
    for (int i = 0; i < 8; ++i) {
        float r = c[i] + bv;
        r = r > 0.0f ? r : 0.0f;
        obase[(size_t)(i + 8 * half) * OUT_FEATS] = r;
    }
}

extern "C" void kernel_launch(void* const* d_in, const int* in_sizes, int n_in,
                              void* d_out, int out_size, void* d_ws, size_t ws_size,
                              hipStream_t stream) {
    const float* feats  = (const float*)d_in[0];
    const int*   src    = (const int*)d_in[1];
    const int*   dst    = (const int*)d_in[2];
    const float* weight = (const float*)d_in[3];
    const float* bias   = (const float*)d_in[4];
    float*       out    = (float*)d_out;

    const int n_nodes = in_sizes[0] / IN_FEATS;   // 50000
    const int n_edges = in_sizes[1];              // 800000

    float* agg = (float*)d_ws;                    // n_nodes*128 f32 = 25.6 MB
    const size_t agg_bytes = (size_t)n_nodes * IN_FEATS * sizeof(float);

    // agg = 0 (graph-capturable memset node)
    hipMemsetAsync(agg, 0, agg_bytes, stream);

    // scatter-sum over edges: one wave per edge, 8 edges per block
    {
        int block = 256;
        int grid  = (n_edges + 7) / 8;            // 100000 blocks
        gc_scatter_kernel<<<grid, block, 0, stream>>>(feats, src, dst, agg, n_edges);
    }

    // GEMM + bias + relu: one block per 16-row M tile (50000 % 16 == 0)
    {
        int grid = n_nodes / 16;                  // 3125
        gc_gemm_relu_kernel<<<grid, 256, 0, stream>>>(agg, weight, bias, out);
    }
}